// InverseStep_90494960926918
// MI455X (gfx1250) — compile-verified
//
#include <hip/hip_runtime.h>
#include <stdint.h>

// ---------------------------------------------------------------------------
// MI455X (gfx1250) implementation.
//
// Problem is store-bandwidth bound: 470 MB of output vs ~17 MB of input and
// ~0.3 GFLOP of math -> floor ~20us at 23.3 TB/s. Strategy:
//   * stage all row data into LDS via CDNA5 async global->LDS copies
//     (global_load_async_to_lds_b32 + s_wait_asynccnt),
//   * compute the 28-wide band sum Mk with V_WMMA_F32_16X16X4_F32
//     (banded-ones A matrix, msq data in B),
//   * emit every output as aligned 16B global_store_b128 (28 floats per
//     (b,r,c) = 7 float4s, 112B stride keeps 16B alignment).
// Global maxima found in a first pass: max distributes over the nonnegative
// product, so max(X) = max_{r,c} mask1[r,c] * max_{b,i} Yg[b,r,c+i] and X is
// never materialized twice.
// ---------------------------------------------------------------------------

#define BANDS 28
#define NCOL  512
#define LROW  539              // NCOL + BANDS - 1
#define NROWS 4096             // 8 * 512
#define XCOUNT 58720256ull     // 8*512*512*28 elements per output tensor

typedef __attribute__((ext_vector_type(2))) float v2f;
typedef __attribute__((ext_vector_type(8))) float v8f;

// ---- CDNA5 async global->LDS copy (ASYNCcnt-tracked, bypasses VGPRs) ------
__device__ __forceinline__ void async_b32(const float* g, const float* lds) {
  // Low 32 bits of the generic address of a __shared__ object are the
  // workgroup-relative LDS byte offset (ISA 10.2 aperture mapping).
  unsigned loff = (unsigned)(unsigned long long)(const void*)lds;
  asm volatile("global_load_async_to_lds_b32 %0, %1, off"
               :: "v"(loff), "v"(g) : "memory");
}
__device__ __forceinline__ void wait_async() {
  asm volatile("s_wait_asynccnt 0x0" ::: "memory");
}

// ---- band sum via V_WMMA_F32_16X16X4_F32 ----------------------------------
// Mk[j] = sum_{i=0..27} msq[j-i]  for j in [0,539)
// Group jb in {0,256,512}: D[m,n] = sum_tau band(m - tau) * msq[jb+16n+tau],
// tau in [-28,16), accumulated in 11 WMMAs of K=4. j = jb + 16n + m.
// A[m,k] = band(m - (tau0+k)) is lane-computable; B[k,n] = msq[jb+16n+tau0+k]
// is a ds_load_b64 (two consecutive floats, 8B aligned since tau0 is even).
__device__ __forceinline__ void yg1_from_wmma(const float* msq_ext /* t=0 at +32 */,
                                              const float* yg, float* yg1, int tid) {
  if (tid < 96) {                    // waves 0..2, full 32-lane EXEC each
    const int lane = tid & 31;
    const int wave = tid >> 5;
    const int jb   = wave * 256;
    const int n    = lane & 15;      // N = lane%16 (B and D)
    const int hi   = lane >> 4;      // half-wave select
    const int m    = n;              // A: M = lane%16
    const int k0   = hi * 2;         // A: K = {0,1} or {2,3}
    v8f acc = {};
#pragma unroll
    for (int it = 0; it < 11; ++it) {
      const int tau0 = -28 + 4 * it;
      const int d0 = m - (tau0 + k0);          // band offset for K=k0
      v2f a;
      a.x = (d0 >= 0 && d0 <= 27) ? 1.0f : 0.0f;
      a.y = (d0 - 1 >= 0 && d0 - 1 <= 27) ? 1.0f : 0.0f;
      const int tB = jb + n * 16 + tau0 + 2 * hi;   // B: K = v + 2*hi
      v2f b;
      b.x = msq_ext[32 + tB];
      b.y = msq_ext[32 + tB + 1];
      acc = __builtin_amdgcn_wmma_f32_16x16x4_f32(
          /*neg_a=*/false, a, /*neg_b=*/false, b,
          /*c_mod=*/(short)0, acc, /*reuse_a=*/false, /*reuse_b=*/false);
    }
    // D layout: lane holds j = jb + 16*(lane%16) + v + 8*hi
#pragma unroll
    for (int v = 0; v < 8; ++v) {
      const int j = jb + n * 16 + v + 8 * hi;
      if (j < LROW) yg1[j] = yg[j] / (acc[v] + 1.0f);  // Yg1 = Yg / (Mk + 1)
    }
  }
}

// ---- stage one (b,r) row into LDS and build yg / yg1 / mask1 --------------
__device__ __forceinline__ void stage_row(const float* __restrict__ inputs,
                                          const float* __restrict__ mask,
                                          int rho, int r, int tid,
                                          float* msq_ext, float* yg,
                                          float* yg1, float* m1) {
  const float* inRow = inputs + (size_t)rho * LROW;   // Yg[b,r,:]
  const float* mRow  = mask + (size_t)rho * NCOL;     // mask[b,r,:]
  const float* m1Row = mask + (size_t)r * NCOL;       // mask[0,r,:]

  // zero guard pads (disjoint from the async target [32,544))
  if (tid < 32) msq_ext[tid] = 0.0f;
  for (int i = 544 + tid; i < 800; i += 256) msq_ext[i] = 0.0f;

  for (int i = tid; i < LROW; i += 256) async_b32(inRow + i, &yg[i]);
  for (int i = tid; i < NCOL; i += 256) async_b32(mRow + i, &msq_ext[32 + i]);
  for (int i = tid; i < NCOL; i += 256) async_b32(m1Row + i, &m1[i]);
  wait_async();
  __syncthreads();

  for (int i = tid; i < NCOL; i += 256) {            // square in place
    float v = msq_ext[32 + i];
    msq_ext[32 + i] = v * v;
  }
  __syncthreads();

  yg1_from_wmma(msq_ext, yg, yg1, tid);
  __syncthreads();
}

// ---- pass 1: global maxima of X and X1 ------------------------------------
__global__ __launch_bounds__(256)
void InverseStep_rowmax_kernel(const float* __restrict__ inputs,
                               const float* __restrict__ mask,
                               unsigned* __restrict__ gmax) {
  __shared__ float msq_ext[800];
  __shared__ float yg[544];
  __shared__ float yg1[544];
  __shared__ float m1[512];
  __shared__ unsigned smax[2];

  const int tid = threadIdx.x;
  const int rho = blockIdx.x;          // b*512 + r
  const int r   = rho & (NCOL - 1);
  if (tid < 2) smax[tid] = 0u;

  stage_row(inputs, mask, rho, r, tid, msq_ext, yg, yg1, m1);

  float cx = 0.0f, c1 = 0.0f;          // all candidates are >= 0
  for (int c = tid; c < NCOL; c += 256) {
    float w = yg[c], w1 = yg1[c];
#pragma unroll
    for (int i = 1; i < BANDS; ++i) {
      w  = fmaxf(w,  yg[c + i]);
      w1 = fmaxf(w1, yg1[c + i]);
    }
    const float mm = m1[c];
    cx = fmaxf(cx, mm * w);
    c1 = fmaxf(c1, mm * w1);
  }
  // nonnegative IEEE floats order identically as uint bit patterns
  atomicMax(&smax[0], __float_as_uint(cx));
  atomicMax(&smax[1], __float_as_uint(c1));
  __syncthreads();
  if (tid == 0) {
    atomicMax(&gmax[0], smax[0]);
    atomicMax(&gmax[1], smax[1]);
  }
}

// ---- pass 2: write X and X1 (store-bandwidth bound) -----------------------
__global__ __launch_bounds__(256)
void InverseStep_write_kernel(const float* __restrict__ inputs,
                              const float* __restrict__ mask,
                              const unsigned* __restrict__ gmax,
                              float* __restrict__ out) {
  __shared__ float msq_ext[800];
  __shared__ float yg[544];
  __shared__ float yg1[544];
  __shared__ float m1[512];

  const int tid = threadIdx.x;
  const int rho = blockIdx.x;
  const int r   = rho & (NCOL - 1);

  stage_row(inputs, mask, rho, r, tid, msq_ext, yg, yg1, m1);

  const float mx  = __uint_as_float(gmax[0]);
  const float mx1 = __uint_as_float(gmax[1]);

  for (int c = tid; c < NCOL; c += 256) {
    const float pX = m1[c] / mx;       // X[..i] = pX * yg[c+i]
    const float p1 = m1[c] / mx1;
    const size_t base = ((size_t)rho * NCOL + c) * BANDS;   // *112B: 16B-aligned
    float4* __restrict__ oX = reinterpret_cast<float4*>(out + base);
    float4* __restrict__ o1 = reinterpret_cast<float4*>(out + XCOUNT + base);
#pragma unroll
    for (int v = 0; v < 7; ++v) {
      float4 a, b;
      a.x = pX * yg[c + 4 * v + 0];
      a.y = pX * yg[c + 4 * v + 1];
      a.z = pX * yg[c + 4 * v + 2];
      a.w = pX * yg[c + 4 * v + 3];
      b.x = p1 * yg1[c + 4 * v + 0];
      b.y = p1 * yg1[c + 4 * v + 1];
      b.z = p1 * yg1[c + 4 * v + 2];
      b.w = p1 * yg1[c + 4 * v + 3];
      oX[v] = a;                       // global_store_b128
      o1[v] = b;
    }
  }
}

__global__ void InverseStep_init_kernel(unsigned* g) {
  if (threadIdx.x < 2) g[threadIdx.x] = 0u;
}

extern "C" void kernel_launch(void* const* d_in, const int* in_sizes, int n_in,
                              void* d_out, int out_size, void* d_ws, size_t ws_size,
                              hipStream_t stream) {
  const float* inputs = (const float*)d_in[0];   // (8,512,539) f32
  const float* mask   = (const float*)d_in[1];   // (8,512,512) f32
  unsigned* gmax = (unsigned*)d_ws;              // [maxX_bits, maxX1_bits]
  float* out = (float*)d_out;                    // X (58.7M) then X1 (58.7M)

  InverseStep_init_kernel<<<1, 32, 0, stream>>>(gmax);
  InverseStep_rowmax_kernel<<<NROWS, 256, 0, stream>>>(inputs, mask, gmax);
  InverseStep_write_kernel<<<NROWS, 256, 0, stream>>>(inputs, mask, gmax, out);
}